// JointLearner_70617852281101
// MI455X (gfx1250) — compile-verified
//
#include <hip/hip_runtime.h>
#include <hip/hip_bf16.h>

#define EMB_D   200
#define ODIM    100
#define NE_SZ   100000
#define NO_SZ   5000
#define NL_SZ   50000
#define BATCH   256
#define OUT_CH  2
#define FW_SZ   10
#define STRIDE_ 5
#define WP_SZ   39      // (200-10)/5 + 1
#define FC_LEN  78      // OUT_CH * WP
#define FC1_LEN 20      // OUT_CH * FW

typedef __attribute__((ext_vector_type(2))) float v2f;
typedef __attribute__((ext_vector_type(8))) float v8f;
typedef __attribute__((address_space(3))) float lds_float;

// 32-bit LDS byte offset of a generic pointer that points into shared memory
__device__ __forceinline__ unsigned lds_offset(const void* p) {
    return (unsigned)(unsigned long long)(lds_float*)p;
}

// async copy of 16 bytes global -> LDS (tracked by ASYNCcnt, no VGPR round-trip)
__device__ __forceinline__ void async_copy_b128(unsigned lds_off, const float* gaddr) {
    asm volatile("global_load_async_to_lds_b128 %0, %1, off"
                 :: "v"(lds_off), "v"(gaddr) : "memory");
}
__device__ __forceinline__ void async_wait0() {
    asm volatile("s_wait_asynccnt 0x0" ::: "memory");
}

// ---------------------------------------------------------------------------
// Kernel 1: per-batch prep.  One block per batch element.
//   s[b]   = ontology_emb[type_idx[b]] + relu(relation_emb[rel2_idx[b]] @ pR_W.T + pR_b)
//   p[b]   = relation * conv(user,1) * conv(day,2) * conv(hour,3) * conv(relu(typeSub@proj_W.T+proj_b),5)
// ---------------------------------------------------------------------------
__global__ __launch_bounds__(256) void prep_kernel(
    const int* __restrict__ rel_idx,  const int* __restrict__ rel2_idx,
    const int* __restrict__ user_idx, const int* __restrict__ hour_idx,
    const int* __restrict__ day_idx,  const int* __restrict__ type_idx,
    const float* __restrict__ entity_emb, const float* __restrict__ ontology_emb,
    const float* __restrict__ relation_emb,
    const float* __restrict__ proj_W, const float* __restrict__ proj_b,
    const float* __restrict__ pR_W,   const float* __restrict__ pR_b,
    const float* __restrict__ fc2_W,  const float* __restrict__ fc2_b,
    const float* __restrict__ fc_W,   const float* __restrict__ fc_b,
    float* __restrict__ s_out, float* __restrict__ p_out)
{
    __shared__ float ts_s[ODIM];         // typeSub row
    __shared__ float e_s[EMB_D];         // current source row (rel2 row first)
    __shared__ float tt0_s[EMB_D];       // relu(typeSub @ proj_W.T + proj_b)
    __shared__ float y_s[FC_LEN];
    __shared__ float k_s[4][FC1_LEN];    // conv kernels for pos {1,2,3,5}

    const int b   = blockIdx.x;
    const int tid = threadIdx.x;

    if (tid < 4 * FC1_LEN) {
        const int posmap[4] = {1, 2, 3, 5};
        int pi = tid / FC1_LEN, i = tid % FC1_LEN;
        k_s[pi][i] = fc2_W[i * 6 + posmap[pi]] + fc2_b[i];
    }
    if (tid < ODIM)  ts_s[tid] = ontology_emb[type_idx[b] * ODIM + tid];
    if (tid < EMB_D) e_s[tid]  = relation_emb[rel2_idx[b] * EMB_D + tid];
    __syncthreads();

    if (tid < ODIM) {                    // subRel + typeSub -> s
        float acc = pR_b[tid];
        const float* wr = &pR_W[tid * EMB_D];
        for (int d = 0; d < EMB_D; ++d) acc = fmaf(e_s[d], wr[d], acc);
        s_out[b * ODIM + tid] = ts_s[tid] + fmaxf(acc, 0.f);
    }
    if (tid < EMB_D) {                   // tt0 = relu(typeSub @ proj_W.T + proj_b)
        float acc = proj_b[tid];
        const float* wr = &proj_W[tid * ODIM];
        for (int o = 0; o < ODIM; ++o) acc = fmaf(ts_s[o], wr[o], acc);
        tt0_s[tid] = fmaxf(acc, 0.f);
    }
    float p_reg = (tid < EMB_D) ? relation_emb[rel_idx[b] * EMB_D + tid] : 0.f;
    __syncthreads();

    // four convolve sources: user(pos1), day(pos2), hour(pos3), tt0(pos5)
    for (int srcI = 0; srcI < 4; ++srcI) {
        if (srcI < 3) {
            const int* ip = (srcI == 0) ? user_idx : (srcI == 1) ? day_idx : hour_idx;
            if (tid < EMB_D) e_s[tid] = entity_emb[(long)ip[b] * EMB_D + tid];
        } else {
            if (tid < EMB_D) e_s[tid] = tt0_s[tid];
        }
        __syncthreads();
        if (tid < FC_LEN) {              // strided 1D conv
            int oc = tid / WP_SZ, wp = tid % WP_SZ;
            const float* kk = &k_s[srcI][oc * FW_SZ];
            const float* er = &e_s[wp * STRIDE_];
            float acc = 0.f;
            #pragma unroll
            for (int f = 0; f < FW_SZ; ++f) acc = fmaf(kk[f], er[f], acc);
            y_s[tid] = acc;
        }
        __syncthreads();
        if (tid < EMB_D) {               // y @ fc_W.T + fc_b, fold into product
            float acc = fc_b[tid];
            const float* wr = &fc_W[tid * FC_LEN];
            #pragma unroll 6
            for (int j = 0; j < FC_LEN; ++j) acc = fmaf(y_s[j], wr[j], acc);
            p_reg *= acc;
        }
        __syncthreads();
    }
    if (tid < EMB_D) p_out[b * EMB_D + tid] = p_reg;
}

// ---------------------------------------------------------------------------
// Kernel 2: ontScores[b,o] = sum_k |s[b,k] - ontology_emb[o,k]|
// 32x32 (b,o) tile per block; s-tile staged via async-to-LDS, o-tile padded
// to stride 101 (conflict-free).
// ---------------------------------------------------------------------------
__global__ __launch_bounds__(256) void ont_kernel(
    const float* __restrict__ s, const float* __restrict__ ontology_emb,
    float* __restrict__ ont_out)
{
    constexpr int BT = 32, OT = 32, OS = 101;
    __shared__ __align__(16) float s_s[BT * ODIM];   // 3200 floats, contiguous
    __shared__ float o_s[OT * OS];

    const int tid = threadIdx.x;
    const int o0 = blockIdx.x * OT;
    const int b0 = blockIdx.y * BT;

    // async-stage the 32 contiguous s rows (12.8 KB) straight into LDS
    {
        const float* gsrc = s + b0 * ODIM;
        for (int i = tid * 4; i < BT * ODIM; i += 256 * 4)
            async_copy_b128(lds_offset(&s_s[i]), gsrc + i);
    }
    for (int i = tid; i < OT * ODIM; i += 256) {
        int r = i / ODIM, c = i % ODIM;
        int orow = o0 + r;
        o_s[r * OS + c] = (orow < NO_SZ) ? ontology_emb[orow * ODIM + c] : 0.f;
    }
    async_wait0();
    __syncthreads();

    for (int q = tid; q < BT * OT; q += 256) {
        int oi = q & 31, bi = q >> 5;
        const float* sr = &s_s[bi * ODIM];
        const float* orr = &o_s[oi * OS];
        float acc = 0.f;
        #pragma unroll 4
        for (int k = 0; k < ODIM; ++k) acc += fabsf(sr[k] - orr[k]);
        int oo = o0 + oi;
        if (oo < NO_SZ)
            __builtin_nontemporal_store(acc, &ont_out[(long)(b0 + bi) * NO_SZ + oo]);
    }
}

// ---------------------------------------------------------------------------
// Kernel 3: fused e5-tile generation + fp32 WMMA GEMM.
// Block handles 32 columns (entities) and full M=256:
//   phase 1: async-stage 32 entity rows into LDS, conv (pos 4) -> y, FC -> e5
//            tile (e5 overwrites the entity buffer, stride 201 = conflict-free)
//   phase 2: 8 waves, each a 32x32 output macro-tile as 2x2 WMMA 16x16 tiles,
//            K=200 in 50 steps of V_WMMA_F32_16X16X4_F32.
// ---------------------------------------------------------------------------
__global__ __launch_bounds__(256) void hype_kernel(
    const float* __restrict__ entity_emb,
    const float* __restrict__ fc2_W, const float* __restrict__ fc2_b,
    const float* __restrict__ fc_W,  const float* __restrict__ fc_b,
    const float* __restrict__ p, float* __restrict__ hype)
{
    constexpr int NT = 32;     // entity columns per block
    constexpr int YS = 80;     // y LDS stride (78 padded)
    constexpr int ES = 201;    // e5 LDS stride (odd mod 64 -> conflict-free)

    __shared__ float y_s[NT * YS];                   // 2560 floats
    __shared__ __align__(16) float ee_s[NT * ES];    // ent (stride 200) then e5 (stride 201)
    __shared__ float k4_s[FC1_LEN];

    const int tid = threadIdx.x;
    const long n_base = (long)blockIdx.x * NT;

    if (tid < FC1_LEN) k4_s[tid] = fc2_W[tid * 6 + 4] + fc2_b[tid];
    // async-stage 32 contiguous entity rows (25.6 KB) straight into LDS
    {
        const float* gsrc = entity_emb + n_base * EMB_D;
        for (int i = tid * 4; i < NT * EMB_D; i += 256 * 4)
            async_copy_b128(lds_offset(&ee_s[i]), gsrc + i);
    }
    async_wait0();
    __syncthreads();

    // conv: y[e, oc*39+wp] = sum_f k4[oc,f] * ent[e, wp*5+f]
    for (int i = tid; i < NT * FC_LEN; i += 256) {
        int e = i / FC_LEN, j = i % FC_LEN;
        int oc = j / WP_SZ, wp = j % WP_SZ;
        const float* er = &ee_s[e * EMB_D + wp * STRIDE_];
        const float* kk = &k4_s[oc * FW_SZ];
        float acc = 0.f;
        #pragma unroll
        for (int f = 0; f < FW_SZ; ++f) acc = fmaf(kk[f], er[f], acc);
        y_s[e * YS + j] = acc;
    }
    __syncthreads();

    // FC: e5[e,d] = fc_b[d] + y[e,:] . fc_W[d,:]  (reads y only -> may overwrite ent)
    for (int i = tid; i < NT * EMB_D; i += 256) {
        int e = i / EMB_D, dd = i % EMB_D;
        float acc = fc_b[dd];
        const float* yr = &y_s[e * YS];
        const float* wr = &fc_W[dd * FC_LEN];
        #pragma unroll 6
        for (int j = 0; j < FC_LEN; ++j) acc = fmaf(yr[j], wr[j], acc);
        ee_s[e * ES + dd] = acc;
    }
    __syncthreads();

    // ---- WMMA GEMM: hype[m, n] = sum_k p[m,k] * e5[n,k] ----
    const int lane = tid & 31;
    const int wave = tid >> 5;
    const int nl   = lane & 15;
    const int hi   = lane >> 4;          // 0: K pair {0,1}, 1: K pair {2,3}
    const int koff = hi * 2;
    const int m0   = wave * 32;          // this wave's 32 output rows

    v8f acc00 = {}, acc01 = {}, acc10 = {}, acc11 = {};
    const float* pr0 = p + (long)(m0 + nl) * EMB_D;
    const float* pr1 = p + (long)(m0 + 16 + nl) * EMB_D;
    __builtin_prefetch(pr0, 0, 0);
    __builtin_prefetch(pr1, 0, 0);
    const float* b0p = &ee_s[nl * ES];
    const float* b1p = &ee_s[(16 + nl) * ES];

    #pragma unroll 5
    for (int k = 0; k < EMB_D; k += 4) {
        v2f a0 = *(const v2f*)(pr0 + k + koff);   // A: 16x4 f32 fragment
        v2f a1 = *(const v2f*)(pr1 + k + koff);
        v2f b0; b0.x = b0p[k + koff]; b0.y = b0p[k + koff + 1];  // B: 4x16 = e5^T
        v2f b1; b1.x = b1p[k + koff]; b1.y = b1p[k + koff + 1];
        acc00 = __builtin_amdgcn_wmma_f32_16x16x4_f32(false, a0, false, b0, (short)0, acc00, false, false);
        acc01 = __builtin_amdgcn_wmma_f32_16x16x4_f32(false, a0, false, b1, (short)0, acc01, false, false);
        acc10 = __builtin_amdgcn_wmma_f32_16x16x4_f32(false, a1, false, b0, (short)0, acc10, false, false);
        acc11 = __builtin_amdgcn_wmma_f32_16x16x4_f32(false, a1, false, b1, (short)0, acc11, false, false);
    }

    // C/D layout: VGPR r -> row base + r (lanes 0-15) / base + 8 + r (lanes 16-31)
    const long n0 = n_base + nl;
    const long n1 = n_base + 16 + nl;
    const int  mr = m0 + hi * 8;
    #pragma unroll
    for (int r = 0; r < 8; ++r) {
        long row0 = (long)(mr + r) * NL_SZ;
        long row1 = (long)(mr + 16 + r) * NL_SZ;
        if (n0 < NL_SZ) {
            __builtin_nontemporal_store(acc00[r], &hype[row0 + n0]);
            __builtin_nontemporal_store(acc10[r], &hype[row1 + n0]);
        }
        if (n1 < NL_SZ) {
            __builtin_nontemporal_store(acc01[r], &hype[row0 + n1]);
            __builtin_nontemporal_store(acc11[r], &hype[row1 + n1]);
        }
    }
}

// ---------------------------------------------------------------------------
extern "C" void kernel_launch(void* const* d_in, const int* in_sizes, int n_in,
                              void* d_out, int out_size, void* d_ws, size_t ws_size,
                              hipStream_t stream) {
    const int*   rel_idx      = (const int*)  d_in[0];
    const int*   rel2_idx     = (const int*)  d_in[1];
    const int*   user_idx     = (const int*)  d_in[2];
    const int*   hour_idx     = (const int*)  d_in[3];
    const int*   day_idx      = (const int*)  d_in[4];
    const int*   type_idx     = (const int*)  d_in[5];
    // d_in[6] = currentBatchSize (unused; B fixed at 256)
    const float* entity_emb   = (const float*)d_in[7];
    const float* ontology_emb = (const float*)d_in[8];
    const float* relation_emb = (const float*)d_in[9];
    const float* proj_W       = (const float*)d_in[10];
    const float* proj_b       = (const float*)d_in[11];
    const float* pR_W         = (const float*)d_in[12];
    const float* pR_b         = (const float*)d_in[13];
    const float* fc2_W        = (const float*)d_in[14];
    const float* fc2_b        = (const float*)d_in[15];
    const float* fc_W         = (const float*)d_in[16];
    const float* fc_b         = (const float*)d_in[17];

    float* ws   = (float*)d_ws;
    float* s_ws = ws;                        // [B, OD]   = 25600 floats
    float* p_ws = ws + BATCH * ODIM;         // [B, D]    = 51200 floats

    float* out      = (float*)d_out;
    float* ont_out  = out;                           // [B, NO]
    float* hype_out = out + (size_t)BATCH * NO_SZ;   // [B, NL]

    prep_kernel<<<BATCH, 256, 0, stream>>>(
        rel_idx, rel2_idx, user_idx, hour_idx, day_idx, type_idx,
        entity_emb, ontology_emb, relation_emb,
        proj_W, proj_b, pR_W, pR_b, fc2_W, fc2_b, fc_W, fc_b,
        s_ws, p_ws);

    ont_kernel<<<dim3((NO_SZ + 31) / 32, BATCH / 32), 256, 0, stream>>>(
        s_ws, ontology_emb, ont_out);

    hype_kernel<<<(NL_SZ + 31) / 32, 256, 0, stream>>>(
        entity_emb, fc2_W, fc2_b, fc_W, fc_b, p_ws, hype_out);
}